// DecoderWithAttention_17205638988438
// MI455X (gfx1250) — compile-verified
//
#include <hip/hip_runtime.h>
#include <hip/hip_bf16.h>
#include <math.h>

// ---- problem constants (match reference) ----
#define Bc    32
#define Tc    64
#define Pc    196          // E*E = 14*14 encoder positions
#define ENCD  2048
#define DECD  512
#define ATTD  512
#define PATCHc 32
#define CHc   3
#define FLATc 3072         // 32*32*3
#define EMBc  256
#define PADc  24           // PATCH/2 + PATCH/4
#define IMGc  512

typedef float v2f __attribute__((ext_vector_type(2)));
typedef float v8f __attribute__((ext_vector_type(8)));

__device__ __forceinline__ float sigm(float x) { return 1.0f / (1.0f + expf(-x)); }

// ============================================================================
// fp32 WMMA GEMM:  Y[M,N] = X[M,K] @ W[N,K]^T + bias  (optional +=Y)
// One wave computes a 32x16 tile (two 16x16 WMMA accumulators sharing one
// B fragment) -> each W row is streamed once per 32 output rows, which is
// optimal L2 traffic for the M=32 recurrent GEMMs.
// Optional row gather for X: row r -> gather[r/group]*gstride + (r%group)*K
// Requires: M multiple of 32; N multiple of 16; K multiple of 64.
// flags bit0: accumulate into existing Y (bias still added).
// ============================================================================
__global__ __launch_bounds__(32) void gemm_wmma_f32(
    const float* __restrict__ X, const float* __restrict__ W,
    const float* __restrict__ bias, float* __restrict__ Y,
    int N, int K,
    const int* __restrict__ gather, int group, long gstride,
    int flags)
{
  const int lane = threadIdx.x;        // 0..31
  const int half = lane >> 4;          // 0 or 1
  const int l15  = lane & 15;
  const int n0 = blockIdx.x * 16;
  const int m0 = blockIdx.y * 32;      // two 16-row tiles

  // A-fragment source rows for this lane
  const int arow0 = m0 + l15;
  const int arow1 = m0 + 16 + l15;
  long abase0, abase1;
  if (gather) {
    abase0 = (long)gather[arow0 / group] * gstride + (long)(arow0 % group) * K;
    abase1 = (long)gather[arow1 / group] * gstride + (long)(arow1 % group) * K;
  } else {
    abase0 = (long)arow0 * K;
    abase1 = (long)arow1 * K;
  }
  const float* __restrict__ a0p = X + abase0;
  const float* __restrict__ a1p = X + abase1;

  // B-fragment: column n = n0 + l15 of W^T  == row (n0+l15) of W
  const float* __restrict__ wrow = W + (long)(n0 + l15) * K;

  const int kk = half * 2;             // lanes 0-15: K offsets 0,1 ; lanes 16-31: 2,3

  v8f acc0 = {0.f, 0.f, 0.f, 0.f, 0.f, 0.f, 0.f, 0.f};
  v8f acc1 = {0.f, 0.f, 0.f, 0.f, 0.f, 0.f, 0.f, 0.f};

  for (int k0 = 0; k0 < K; k0 += 64) {
    // pull the next chunk toward the WGP while we compute on this one
    __builtin_prefetch(wrow + k0 + 64, 0, 1);
    __builtin_prefetch(a0p  + k0 + 64, 0, 1);
    __builtin_prefetch(a1p  + k0 + 64, 0, 1);
#pragma unroll
    for (int k = k0; k < k0 + 64; k += 4) {
      v2f b  = *(const v2f*)(wrow + k + kk);    // B 4x16 fragment (col n, K-major)
      v2f a0 = *(const v2f*)(a0p  + k + kk);    // A 16x4 fragments
      v2f a1 = *(const v2f*)(a1p  + k + kk);
      acc0 = __builtin_amdgcn_wmma_f32_16x16x4_f32(
          false, a0, false, b, (short)0, acc0, false, false);
      acc1 = __builtin_amdgcn_wmma_f32_16x16x4_f32(
          false, a1, false, b, (short)0, acc1, false, false);
    }
  }

  // C/D layout: VGPR v -> (row base+v, col n0+lane) lanes 0-15,
  //             (row base+v+8, col n0+lane-16) lanes 16-31.
  const int col = n0 + l15;
  const float bval = bias ? bias[col] : 0.f;
#pragma unroll
  for (int v = 0; v < 8; ++v) {
    const int row0 = m0 + v + half * 8;
    const long off0 = (long)row0 * N + col;
    float r0 = acc0[v] + bval;
    if (flags & 1) r0 += Y[off0];
    Y[off0] = r0;

    const int row1 = m0 + 16 + v + half * 8;
    const long off1 = (long)row1 * N + col;
    float r1 = acc1[v] + bval;
    if (flags & 1) r1 += Y[off1];
    Y[off1] = r1;
  }
}

// ============================================================================
// Stable argsort of -lengths (B=32), writes sorted lengths / sort_ind outputs
// and gathers sequence_offset into sorted order.
// ============================================================================
__global__ __launch_bounds__(32) void k_sort(
    const int* __restrict__ cap_len, const float* __restrict__ seq_off,
    int* __restrict__ sortind, int* __restrict__ len_s,
    float* __restrict__ o_len, float* __restrict__ o_sortind,
    float* __restrict__ o_seqoff)
{
  const int i = threadIdx.x;           // 0..31
  const int li = cap_len[i];
  int rank = 0;
  for (int j = 0; j < Bc; ++j) {
    const int lj = cap_len[j];
    if (lj > li || (lj == li && j < i)) rank++;
  }
  sortind[rank] = i;
  __syncthreads();
  const int src = sortind[i];
  len_s[i] = cap_len[src];
  o_len[i] = (float)cap_len[src];
  o_sortind[i] = (float)src;
  for (int k = 0; k < Tc * 2; ++k)
    o_seqoff[i * Tc * 2 + k] = seq_off[src * Tc * 2 + k];
}

// ============================================================================
// Patch extraction (sorted order) with implicit zero padding (PAD=24).
// patches[b*T+t][c*1024 + py*32 + px]
// ============================================================================
__global__ __launch_bounds__(256) void k_patches(
    const float* __restrict__ imgs, const int* __restrict__ seq,
    const int* __restrict__ sortind, float* __restrict__ patches)
{
  const int bt = blockIdx.x;
  const int b = bt / Tc, t = bt % Tc;
  const int sb = sortind[b];
  const int kx = seq[((long)sb * Tc + t) * 2 + 0];
  const int ky = seq[((long)sb * Tc + t) * 2 + 1];
  const float* __restrict__ img = imgs + (long)sb * CHc * IMGc * IMGc;
  float* __restrict__ dst = patches + (long)bt * FLATc;
  for (int i = threadIdx.x; i < FLATc; i += blockDim.x) {
    const int c = i / (PATCHc * PATCHc);
    const int rem = i % (PATCHc * PATCHc);
    const int py = rem / PATCHc, px = rem % PATCHc;
    const int yy = ky + py - PADc;
    const int xx = kx + px - PADc;
    float v = 0.f;
    if (yy >= 0 && yy < IMGc && xx >= 0 && xx < IMGc)
      v = img[((long)c * IMGc + yy) * IMGc + xx];
    dst[i] = v;
  }
}

// mean over 196 positions of sorted encoder features -> (B, 2048)
__global__ __launch_bounds__(256) void k_encmean(
    const float* __restrict__ enc, const int* __restrict__ sortind,
    float* __restrict__ mean)
{
  const int b = blockIdx.x;
  const int d = blockIdx.y * blockDim.x + threadIdx.x;   // 0..2047
  const float* __restrict__ src = enc + (long)sortind[b] * Pc * ENCD + d;
  float s = 0.f;
  for (int p = 0; p < Pc; ++p) s += src[(long)p * ENCD];
  mean[b * ENCD + d] = s * (1.0f / (float)Pc);
}

// init_in = concat(enc_mean, emb[:,0,:]) -> (B, 2304)
__global__ __launch_bounds__(256) void k_initin(
    const float* __restrict__ mean, const float* __restrict__ emb,
    float* __restrict__ init_in)
{
  const int b = blockIdx.x;
  for (int i = threadIdx.x; i < ENCD + EMBc; i += blockDim.x)
    init_in[b * (ENCD + EMBc) + i] =
        (i < ENCD) ? mean[b * ENCD + i]
                   : emb[((long)b * Tc + 0) * EMBc + (i - ENCD)];
}

// ============================================================================
// Attention scores + softmax: one block per batch element.
// Each wave handles one position p with lanes striding the 512-dim row
// (coalesced 128B accesses) + shuffle reduction.
// Writes normalized alpha to ws (for awe) and masked alpha to output.
// ============================================================================
__global__ __launch_bounds__(256) void k_attention(
    const float* __restrict__ att1, const float* __restrict__ att2,
    const float* __restrict__ w_full, const float* __restrict__ b_full,
    const int* __restrict__ len_s,
    float* __restrict__ alpha_ws, float* __restrict__ o_alpha, int t)
{
  __shared__ float s_att2[ATTD];
  __shared__ float s_sc[Pc];
  __shared__ float s_max, s_sum;
  const int b = blockIdx.x;
  const int wave = threadIdx.x >> 5;   // 0..7
  const int lane = threadIdx.x & 31;

  for (int i = threadIdx.x; i < ATTD; i += blockDim.x) s_att2[i] = att2[b * ATTD + i];
  __syncthreads();

  const float* __restrict__ a1 = att1 + (long)b * Pc * ATTD;
  const float bf = b_full[0];
  for (int p = wave; p < Pc; p += 8) {
    const float* __restrict__ row = a1 + (long)p * ATTD;
    float s = 0.f;
    for (int a = lane; a < ATTD; a += 32) {
      float v = row[a] + s_att2[a];
      v = v > 0.f ? v : 0.f;
      s += v * w_full[a];
    }
    for (int off = 16; off > 0; off >>= 1) s += __shfl_down(s, off);
    if (lane == 0) s_sc[p] = s + bf;
  }
  __syncthreads();
  if (threadIdx.x == 0) {
    float m = -INFINITY;
    for (int p = 0; p < Pc; ++p) m = fmaxf(m, s_sc[p]);
    s_max = m;
  }
  __syncthreads();
  for (int p = threadIdx.x; p < Pc; p += blockDim.x) s_sc[p] = expf(s_sc[p] - s_max);
  __syncthreads();
  if (threadIdx.x == 0) {
    float s = 0.f;
    for (int p = 0; p < Pc; ++p) s += s_sc[p];
    s_sum = s;
  }
  __syncthreads();
  const float inv = 1.0f / s_sum;
  const bool mask = t < len_s[b];
  for (int p = threadIdx.x; p < Pc; p += blockDim.x) {
    const float al = s_sc[p] * inv;
    alpha_ws[b * Pc + p] = al;
    o_alpha[((long)b * Tc + t) * Pc + p] = mask ? al : 0.f;
  }
}

// awe[b,d] = sum_p enc_sorted[b,p,d] * alpha[b,p]
// Grid (B, ENCD/256): 256 blocks streaming the 51MB encoder tensor from L2.
__global__ __launch_bounds__(256) void k_awe(
    const float* __restrict__ enc, const int* __restrict__ sortind,
    const float* __restrict__ alpha_ws, float* __restrict__ awe)
{
  __shared__ float s_al[Pc];
  const int b = blockIdx.x;
  const int d = blockIdx.y * blockDim.x + threadIdx.x;
  for (int p = threadIdx.x; p < Pc; p += blockDim.x) s_al[p] = alpha_ws[b * Pc + p];
  __syncthreads();
  const float* __restrict__ es = enc + (long)sortind[b] * Pc * ENCD + d;
  float s = 0.f;
  for (int p = 0; p < Pc; ++p) s += es[(long)p * ENCD] * s_al[p];
  awe[b * ENCD + d] = s;
}

// x = concat(sigmoid(gatepre) * awe, emb[:, t, :])  -> (B, 2304)
__global__ __launch_bounds__(256) void k_buildx(
    const float* __restrict__ gatepre, const float* __restrict__ awe,
    const float* __restrict__ emb, float* __restrict__ x, int t)
{
  const int b = blockIdx.x;
  for (int i = threadIdx.x; i < ENCD + EMBc; i += blockDim.x) {
    float v;
    if (i < ENCD) v = sigm(gatepre[b * ENCD + i]) * awe[b * ENCD + i];
    else          v = emb[((long)b * Tc + t) * EMBc + (i - ENCD)];
    x[b * (ENCD + EMBc) + i] = v;
  }
}

// ============================================================================
// LSTM pointwise + tiny output heads (N=2 and N=1 dots done with VALU reduce).
// One block per batch element. Updates h,c under length mask.
// ============================================================================
__global__ __launch_bounds__(256) void k_lstm(
    const float* __restrict__ gates, float* __restrict__ h, float* __restrict__ c,
    const int* __restrict__ len_s,
    const float* __restrict__ W_fc, const float* __restrict__ b_fc,
    const float* __restrict__ W_stop, const float* __restrict__ b_stop,
    float* __restrict__ o_pred, float* __restrict__ o_stop, int t)
{
  __shared__ float s_h[DECD];
  __shared__ float r0[8], r1[8], r2[8];
  const int b = blockIdx.x;
  const bool mask = t < len_s[b];
  const long gb = (long)b * 4 * DECD;

  for (int j = threadIdx.x; j < DECD; j += blockDim.x) {
    const float gi = gates[gb + j];
    const float gf = gates[gb + DECD + j];
    const float gg = gates[gb + 2 * DECD + j];
    const float go = gates[gb + 3 * DECD + j];
    const float cn = sigm(gf) * c[b * DECD + j] + sigm(gi) * tanhf(gg);
    const float hn = sigm(go) * tanhf(cn);
    s_h[j] = hn;
    if (mask) { c[b * DECD + j] = cn; h[b * DECD + j] = hn; }
  }
  __syncthreads();

  float l0 = 0.f, l1 = 0.f, l2 = 0.f;
  for (int j = threadIdx.x; j < DECD; j += blockDim.x) {
    const float hv = s_h[j];
    l0 += hv * W_fc[j];
    l1 += hv * W_fc[DECD + j];
    l2 += hv * W_stop[j];
  }
  for (int off = 16; off > 0; off >>= 1) {
    l0 += __shfl_down(l0, off);
    l1 += __shfl_down(l1, off);
    l2 += __shfl_down(l2, off);
  }
  if ((threadIdx.x & 31) == 0) {
    const int w = threadIdx.x >> 5;
    r0[w] = l0; r1[w] = l1; r2[w] = l2;
  }
  __syncthreads();
  if (threadIdx.x == 0) {
    float p0 = 0.f, p1 = 0.f, p2 = 0.f;
    for (int w = 0; w < (int)(blockDim.x >> 5); ++w) { p0 += r0[w]; p1 += r1[w]; p2 += r2[w]; }
    p0 += b_fc[0]; p1 += b_fc[1]; p2 += b_stop[0];
    o_pred[((long)b * Tc + t) * 2 + 0] = mask ? p0 : 0.f;
    o_pred[((long)b * Tc + t) * 2 + 1] = mask ? p1 : 0.f;
    o_stop[(long)b * Tc + t] = mask ? sigm(p2) : 0.f;
  }
}

// ============================================================================
// Host-side orchestration
// ============================================================================
extern "C" void kernel_launch(void* const* d_in, const int* in_sizes, int n_in,
                              void* d_out, int out_size, void* d_ws, size_t ws_size,
                              hipStream_t stream) {
  (void)in_sizes; (void)n_in; (void)out_size; (void)ws_size;

  const float* encoder_out = (const float*)d_in[0];
  const float* imgs        = (const float*)d_in[1];
  const int*   sequence    = (const int*)  d_in[2];
  const float* seq_off     = (const float*)d_in[3];
  const int*   cap_len     = (const int*)  d_in[4];
  const float* W_pe        = (const float*)d_in[5];
  const float* b_pe        = (const float*)d_in[6];
  const float* W_enc_att   = (const float*)d_in[7];
  const float* b_enc_att   = (const float*)d_in[8];
  const float* W_dec_att   = (const float*)d_in[9];
  const float* b_dec_att   = (const float*)d_in[10];
  const float* w_full      = (const float*)d_in[11];
  const float* b_full      = (const float*)d_in[12];
  const float* W_init_h    = (const float*)d_in[13];
  const float* b_init_h    = (const float*)d_in[14];
  const float* W_init_c    = (const float*)d_in[15];
  const float* b_init_c    = (const float*)d_in[16];
  const float* W_fb        = (const float*)d_in[17];
  const float* b_fb        = (const float*)d_in[18];
  const float* W_ih        = (const float*)d_in[19];
  const float* b_ih        = (const float*)d_in[20];
  const float* W_hh        = (const float*)d_in[21];
  const float* b_hh        = (const float*)d_in[22];
  const float* W_fc        = (const float*)d_in[23];
  const float* b_fc        = (const float*)d_in[24];
  const float* W_stop      = (const float*)d_in[25];
  const float* b_stop      = (const float*)d_in[26];

  // ---- workspace layout ----
  int*   sortind = (int*)d_ws;                 // 32
  int*   len_s   = sortind + 32;               // 32
  float* wsf     = (float*)d_ws + 64;
  float* patches = wsf;                        // 2048*3072
  float* emb     = patches + (long)Bc * Tc * FLATc;      // 2048*256
  float* att1    = emb + (long)Bc * Tc * EMBc;           // 6272*512
  float* encmean = att1 + (long)Bc * Pc * ATTD;          // 32*2048
  float* init_in = encmean + Bc * ENCD;                  // 32*2304
  float* h       = init_in + Bc * (ENCD + EMBc);         // 32*512
  float* c       = h + Bc * DECD;                        // 32*512
  float* att2    = c + Bc * DECD;                        // 32*512
  float* awe     = att2 + Bc * ATTD;                     // 32*2048
  float* gatepre = awe + Bc * ENCD;                      // 32*2048
  float* x       = gatepre + Bc * ENCD;                  // 32*2304
  float* gates   = x + Bc * (ENCD + EMBc);               // 32*2048
  float* alpha_ws= gates + Bc * 4 * DECD;                // 32*196

  // ---- output layout (concatenated tuple, float32) ----
  float* out       = (float*)d_out;
  float* o_pred    = out;                           // B*T*2   = 4096
  float* o_stop    = out + Bc * Tc * 2;             // B*T     = 2048
  float* o_seqoff  = o_stop + Bc * Tc;              // B*T*2   = 4096
  float* o_len     = o_seqoff + Bc * Tc * 2;        // B       = 32
  float* o_alpha   = o_len + Bc;                    // B*T*196 = 401408
  float* o_sortind = o_alpha + (long)Bc * Tc * Pc;  // B       = 32

  // ---- setup phase ----
  k_sort<<<1, 32, 0, stream>>>(cap_len, seq_off, sortind, len_s, o_len, o_sortind, o_seqoff);
  k_patches<<<Bc * Tc, 256, 0, stream>>>(imgs, sequence, sortind, patches);

  // emb = patches @ W_pe^T + b_pe   (M=2048, N=256, K=3072)
  gemm_wmma_f32<<<dim3(EMBc / 16, (Bc * Tc) / 32), 32, 0, stream>>>(
      patches, W_pe, b_pe, emb, EMBc, FLATc, nullptr, 1, 0, 0);

  k_encmean<<<dim3(Bc, ENCD / 256), 256, 0, stream>>>(encoder_out, sortind, encmean);
  k_initin<<<Bc, 256, 0, stream>>>(encmean, emb, init_in);

  // h = init_in @ W_init_h^T + b_init_h ; c likewise (M=32, N=512, K=2304)
  gemm_wmma_f32<<<dim3(DECD / 16, 1), 32, 0, stream>>>(
      init_in, W_init_h, b_init_h, h, DECD, ENCD + EMBc, nullptr, 1, 0, 0);
  gemm_wmma_f32<<<dim3(DECD / 16, 1), 32, 0, stream>>>(
      init_in, W_init_c, b_init_c, c, DECD, ENCD + EMBc, nullptr, 1, 0, 0);

  // att1 = enc_sorted @ W_enc_att^T + b_enc_att (M=6272, N=512, K=2048, gathered rows)
  gemm_wmma_f32<<<dim3(ATTD / 16, (Bc * Pc) / 32), 32, 0, stream>>>(
      encoder_out, W_enc_att, b_enc_att, att1, ATTD, ENCD,
      sortind, Pc, (long)Pc * ENCD, 0);

  // ---- recurrent phase ----
  for (int t = 0; t < Tc; ++t) {
    // att2 = h @ W_dec_att^T + b_dec_att (M=32, N=512, K=512)
    gemm_wmma_f32<<<dim3(ATTD / 16, 1), 32, 0, stream>>>(
        h, W_dec_att, b_dec_att, att2, ATTD, DECD, nullptr, 1, 0, 0);

    k_attention<<<Bc, 256, 0, stream>>>(att1, att2, w_full, b_full,
                                        len_s, alpha_ws, o_alpha, t);
    k_awe<<<dim3(Bc, ENCD / 256), 256, 0, stream>>>(encoder_out, sortind,
                                                    alpha_ws, awe);

    // gatepre = h @ W_fb^T + b_fb (M=32, N=2048, K=512)
    gemm_wmma_f32<<<dim3(ENCD / 16, 1), 32, 0, stream>>>(
        h, W_fb, b_fb, gatepre, ENCD, DECD, nullptr, 1, 0, 0);

    k_buildx<<<Bc, 256, 0, stream>>>(gatepre, awe, emb, x, t);

    // gates = x @ W_ih^T + b_ih (M=32, N=2048, K=2304)
    gemm_wmma_f32<<<dim3((4 * DECD) / 16, 1), 32, 0, stream>>>(
        x, W_ih, b_ih, gates, 4 * DECD, ENCD + EMBc, nullptr, 1, 0, 0);
    // gates += h @ W_hh^T + b_hh (M=32, N=2048, K=512, accumulate)
    gemm_wmma_f32<<<dim3((4 * DECD) / 16, 1), 32, 0, stream>>>(
        h, W_hh, b_hh, gates, 4 * DECD, DECD, nullptr, 1, 0, 1);

    k_lstm<<<Bc, 256, 0, stream>>>(gates, h, c, len_s,
                                   W_fc, b_fc, W_stop, b_stop,
                                   o_pred, o_stop, t);
  }
}